// Prompt_4913442586869
// MI455X (gfx1250) — compile-verified
//
#include <hip/hip_runtime.h>
#include <hip/hip_bf16.h>

#define B_    4096
#define POOL_ 1024
#define SEL_  5
#define PLEN_ 5
#define DIM_  768
#define EPS_  1e-8f

typedef __attribute__((ext_vector_type(2))) float v2f;
typedef __attribute__((ext_vector_type(8))) float v8f;

// ---------------------------------------------------------------------------
// Kernel 1: row L2-norms of query (rows 0..B-1) and prompt_key (rows B..B+POOL-1)
// ---------------------------------------------------------------------------
__global__ void prompt_norm_kernel(const float* __restrict__ q,
                                   const float* __restrict__ k,
                                   float* __restrict__ qn,
                                   float* __restrict__ kn)
{
    __shared__ float red[8];
    const int row = blockIdx.x;
    const float* src;
    float* dst;
    if (row < B_) { src = q + (size_t)row * DIM_;        dst = qn + row; }
    else          { src = k + (size_t)(row - B_) * DIM_; dst = kn + (row - B_); }

    float s = 0.0f;
    for (int i = threadIdx.x; i < DIM_; i += blockDim.x) {
        float v = src[i];
        s += v * v;
    }
    // wave32 reduction
    #pragma unroll
    for (int off = 16; off > 0; off >>= 1)
        s += __shfl_down(s, off, 32);
    const int lane = threadIdx.x & 31;
    const int wave = threadIdx.x >> 5;
    if (lane == 0) red[wave] = s;
    __syncthreads();
    if (threadIdx.x == 0) {
        float t = 0.0f;
        #pragma unroll
        for (int w = 0; w < 8; ++w) t += red[w];
        *dst = sqrtf(t);
    }
}

// ---------------------------------------------------------------------------
// Kernel 2: dots = query @ key^T via V_WMMA_F32_16X16X4_F32, fused cosine
// epilogue: match = 1 - dots * rcp(max(qn*kn, eps)). One wave32 per 16x16 tile.
//
// A (16x4 f32): lanes 0-15 -> rows, VGPR0=K0,VGPR1=K1; lanes 16-31 -> K2,K3.
// B (4x16 f32): B = key^T, so B-fragment loads are symmetric to A from key.
// C (16x16 f32, 8 VGPRs): VGPR g -> row bm+g+8*(lane>=16), col bn+(lane&15).
// ---------------------------------------------------------------------------
__global__ void prompt_match_wmma_kernel(const float* __restrict__ q,
                                         const float* __restrict__ k,
                                         const float* __restrict__ qn,
                                         const float* __restrict__ kn,
                                         float* __restrict__ match)
{
    const int lane = threadIdx.x & 31;
    const int wave = threadIdx.x >> 5;                 // 4 waves per block, tiled in N
    const int bn = (blockIdx.x * 4 + wave) * 16;       // POOL tile
    const int bm = blockIdx.y * 16;                    // B tile

    const int half = lane >> 4;                        // 0: K0..K1, 1: K2..K3
    const int r    = lane & 15;

    const float* qrow = q + (size_t)(bm + r) * DIM_ + half * 2;
    const float* krow = k + (size_t)(bn + r) * DIM_ + half * 2;

    v8f c = {};
    #pragma unroll 4
    for (int k0 = 0; k0 < DIM_; k0 += 4) {
        v2f a = *(const v2f*)(qrow + k0);
        v2f b = *(const v2f*)(krow + k0);
        c = __builtin_amdgcn_wmma_f32_16x16x4_f32(
                /*neg_a=*/false, a, /*neg_b=*/false, b,
                /*c_mod=*/(short)0, c, /*reuse_a=*/false, /*reuse_b=*/false);
    }

    const int col = bn + r;
    const float knv = kn[col];
    #pragma unroll
    for (int g = 0; g < 8; ++g) {
        const int row = bm + g + half * 8;
        // hardware v_rcp_f32 (~1 ulp): fine for similarity ranking, avoids the
        // full IEEE div_scale/div_fixup sequence (10 VALU ops -> 2).
        const float rdenom = __builtin_amdgcn_rcpf(fmaxf(qn[row] * knv, EPS_));
        match[(size_t)row * POOL_ + col] = fmaf(-c[g], rdenom, 1.0f);
    }
}

// ---------------------------------------------------------------------------
// Kernel 3: per-row top-5 smallest (ascending, lowest index wins ties),
// matching jax.lax.top_k(-match). One thread per row.
// ---------------------------------------------------------------------------
__global__ void prompt_topk_kernel(const float* __restrict__ match,
                                   float* __restrict__ sim,
                                   int* __restrict__ idx)
{
    const int b = blockIdx.x * blockDim.x + threadIdx.x;
    if (b >= B_) return;

    float bv[SEL_];
    int   bi[SEL_];
    #pragma unroll
    for (int s = 0; s < SEL_; ++s) { bv[s] = __builtin_inff(); bi[s] = 0; }

    const float* m = match + (size_t)b * POOL_;
    for (int c = 0; c < POOL_; ++c) {
        const float v = m[c];
        if (v < bv[SEL_ - 1]) {
            int j = SEL_ - 1;
            while (j > 0 && v < bv[j - 1]) {
                bv[j] = bv[j - 1];
                bi[j] = bi[j - 1];
                --j;
            }
            bv[j] = v;
            bi[j] = c;
        }
    }
    #pragma unroll
    for (int s = 0; s < SEL_; ++s) {
        sim[b * SEL_ + s] = bv[s];
        idx[b * SEL_ + s] = bi[s];
    }
}

// ---------------------------------------------------------------------------
// Kernel 4: selection[b,s] = prompts[idx[b,s]]  (315 MB stream; float4 copies)
// One block per (b,s): 5*768 floats = 960 float4.
// ---------------------------------------------------------------------------
__global__ void prompt_gather_kernel(const float* __restrict__ prompts,
                                     const int* __restrict__ idx,
                                     float* __restrict__ sel)
{
    const int t = blockIdx.x;                 // 0 .. B*SEL-1
    const int p = idx[t];
    const float4* __restrict__ src = (const float4*)(prompts + (size_t)p * PLEN_ * DIM_);
    float4* __restrict__ dst       = (float4*)(sel + (size_t)t * PLEN_ * DIM_);
    const int n4 = (PLEN_ * DIM_) / 4;        // 960
    for (int i = threadIdx.x; i < n4; i += blockDim.x)
        dst[i] = src[i];
}

// ---------------------------------------------------------------------------
extern "C" void kernel_launch(void* const* d_in, const int* in_sizes, int n_in,
                              void* d_out, int out_size, void* d_ws, size_t ws_size,
                              hipStream_t stream)
{
    const float* query   = (const float*)d_in[0];   // [B, DIM]
    const float* key     = (const float*)d_in[1];   // [POOL, DIM]
    const float* prompts = (const float*)d_in[2];   // [POOL, PLEN, DIM]

    // workspace layout
    float* qn    = (float*)d_ws;                            // [B]
    float* kn    = qn + B_;                                 // [POOL]
    float* match = kn + POOL_;                              // [B, POOL]
    int*   idx   = (int*)(match + (size_t)B_ * POOL_);      // [B, SEL]

    // outputs: similarity [B, SEL] then selection [B, SEL, PLEN, DIM]
    float* sim = (float*)d_out;
    float* sel = sim + (size_t)B_ * SEL_;

    // 1) norms
    prompt_norm_kernel<<<B_ + POOL_, 256, 0, stream>>>(query, key, qn, kn);

    // 2) WMMA GEMM + cosine epilogue: grid (POOL/16/4, B/16), 4 waves/block
    dim3 g2(POOL_ / 16 / 4, B_ / 16);
    prompt_match_wmma_kernel<<<g2, 128, 0, stream>>>(query, key, qn, kn, match);

    // 3) top-5 per row
    prompt_topk_kernel<<<(B_ + 255) / 256, 256, 0, stream>>>(match, sim, idx);

    // 4) gather selection
    prompt_gather_kernel<<<B_ * SEL_, 256, 0, stream>>>(prompts, idx, sel);
}